// CausalDepthwiseConv1d_2044404433424
// MI455X (gfx1250) — compile-verified
//
#include <hip/hip_runtime.h>
#include <cstdint>

// Causal depthwise conv1d, fp32:
//   B=8, C=1024, T=8192, K=4, dil=1, left-pad 3.
//   y[b,c,t] = bias[c] + sum_k w[c,0,k] * x[b,c,t-3+k]   (x[t<0] = 0)
//
// Roofline: 0.54 GFLOP vs 536 MB minimum traffic -> ~1 FLOP/byte ->
// pure HBM-bandwidth kernel (~23 us at 23.3 TB/s). Strategy:
//  - one block per (b,c) row (8192 blocks, 64K waves)
//  - stage the 32KB row into LDS with CDNA5 async global->LDS b128 copies
//    (ASYNCcnt path, bypasses VGPR writeback), causal pad = 4 zeros in LDS
//    => exactly 1.0x read traffic, no halo amplification
//  - all streaming traffic is single-use (no reuse, working set >> L2), so
//    loads use TH_LOAD_NT and stores use non-temporal b128 stores.

#define B_DIM 8
#define C_DIM 1024
#define T_DIM 8192
#define THREADS 256
#define VECS_PER_THREAD (T_DIM / (THREADS * 4)) // 8 x b128 per thread

typedef __attribute__((ext_vector_type(4))) float f32x4;

__global__ __launch_bounds__(THREADS) void CausalDepthwiseConv1d_kernel(
    const float* __restrict__ x,   // (B, C, T)
    const float* __restrict__ w,   // (C, 1, K)
    const float* __restrict__ bia, // (C,)
    float* __restrict__ out)       // (B, C, T)
{
    // tile[0..3] = causal zero pad; tile[4 + j] = x[row, j]
    __shared__ float tile[T_DIM + 4];

    const int row = blockIdx.x;            // 0 .. B*C-1
    const int c   = row & (C_DIM - 1);     // channel (block-uniform -> s_load)
    const float* __restrict__ xrow = x + (size_t)row * T_DIM;
    float* __restrict__ orow       = out + (size_t)row * T_DIM;

    // Zero the 4-float left halo (the causal pad).
    if (threadIdx.x == 0) {
        tile[0] = 0.0f; tile[1] = 0.0f; tile[2] = 0.0f; tile[3] = 0.0f;
    }

    // Async-copy the row into LDS: 8 x b128 per thread, coalesced
    // (consecutive lanes -> consecutive 16B chunks each iteration).
    // Non-temporal: each input byte is consumed exactly once.
#pragma unroll
    for (int i = 0; i < VECS_PER_THREAD; ++i) {
        const int e = (i * THREADS + (int)threadIdx.x) * 4; // float index in row
        // Low 32 bits of a generic pointer into LDS == wave-relative LDS byte
        // address (aperture lives in the high bits), which is what the async
        // op's LDS-address VGPR expects.
        unsigned lds_off = (unsigned)(uintptr_t)(&tile[4 + e]);
        const float* gaddr = xrow + e;
        asm volatile("global_load_async_to_lds_b128 %0, %1, off th:TH_LOAD_NT"
                     :
                     : "v"(lds_off), "v"(gaddr)
                     : "memory");
    }
    // Wait for this wave's async LDS writes, then make them visible
    // workgroup-wide.
    asm volatile("s_wait_asynccnt 0" ::: "memory");
    __syncthreads();

    // Block-uniform filter taps + bias -> scalar loads.
    const float w0 = w[c * 4 + 0];
    const float w1 = w[c * 4 + 1];
    const float w2 = w[c * 4 + 2];
    const float w3 = w[c * 4 + 3];
    const float bv = bia[c];

    // Compute 4 outputs per pair of b128 LDS reads, stream results to HBM
    // with non-temporal b128 stores (output is written once, never re-read).
#pragma unroll
    for (int i = 0; i < VECS_PER_THREAD; ++i) {
        const int e = (i * THREADS + (int)threadIdx.x) * 4;
        // p = x[e-4 .. e-1] (tile[e .. e+3]), v = x[e .. e+3]
        const f32x4 p = *(const f32x4*)(&tile[e]);
        const f32x4 v = *(const f32x4*)(&tile[e + 4]);
        f32x4 o;
        o.x = fmaf(w0, p.y, fmaf(w1, p.z, fmaf(w2, p.w, fmaf(w3, v.x, bv))));
        o.y = fmaf(w0, p.z, fmaf(w1, p.w, fmaf(w2, v.x, fmaf(w3, v.y, bv))));
        o.z = fmaf(w0, p.w, fmaf(w1, v.x, fmaf(w2, v.y, fmaf(w3, v.z, bv))));
        o.w = fmaf(w0, v.x, fmaf(w1, v.y, fmaf(w2, v.z, fmaf(w3, v.w, bv))));
        __builtin_nontemporal_store(o, (f32x4*)(&orow[e]));
    }
}

extern "C" void kernel_launch(void* const* d_in, const int* in_sizes, int n_in,
                              void* d_out, int out_size, void* d_ws, size_t ws_size,
                              hipStream_t stream) {
    (void)in_sizes; (void)n_in; (void)out_size; (void)d_ws; (void)ws_size;
    const float* x = (const float*)d_in[0];  // (8, 1024, 8192) fp32
    const float* w = (const float*)d_in[1];  // (1024, 1, 4)    fp32
    const float* b = (const float*)d_in[2];  // (1024,)         fp32
    float* out     = (float*)d_out;          // (8, 1024, 8192) fp32

    dim3 grid(B_DIM * C_DIM);   // one block per (batch, channel) row
    dim3 block(THREADS);
    CausalDepthwiseConv1d_kernel<<<grid, block, 0, stream>>>(x, w, b, out);
}